// GraphNetwork_59399397704022
// MI455X (gfx1250) — compile-verified
//
#include <hip/hip_runtime.h>

typedef float v2f __attribute__((ext_vector_type(2)));
typedef float v8f __attribute__((ext_vector_type(8)));

#define NEG_SLOPE 0.2f

// ---------------------------------------------------------------------------
// Fused dual GEMM with fp32 WMMA:  H = X @ W ;  R = X @ Wres + bias
// X: [N,128] row-major, W/Wres: [128,C] row-major, H/R: [N,C].
// One wave per 16x64 output strip: 4 column tiles x {W, Wres} = 8 accumulators,
// sharing one A fragment per k-step -> 8 v_wmma_f32_16x16x4_f32 per A load.
// Fragment layouts (ISA 32-bit WMMA tables):
//   A (16x4):  lane%16 = M, K = kb + 2*(lane/16) + v
//   B (4x16):  lane%16 = N, K = kb + 2*(lane/16) + v
//   C/D:       lane%16 = col, row = v + 8*(lane/16)
// C is a template parameter so all strides fold into immediate offsets.
// ---------------------------------------------------------------------------
template <int C>
__global__ __launch_bounds__(32) void gat_gemm_dual_wmma(
    const float* __restrict__ X,
    const float* __restrict__ W,
    const float* __restrict__ Wr,
    const float* __restrict__ bias,
    float* __restrict__ H,
    float* __restrict__ R)
{
    const int lane = threadIdx.x;      // 0..31
    const int m    = lane & 15;
    const int half = lane >> 4;        // 0 or 1
    const int r0   = blockIdx.x * 16;
    const int c0   = blockIdx.y * 64;  // 4 adjacent 16-wide column tiles

    // Per-lane base pointers; all loop/epilogue offsets are compile-time.
    const float* __restrict__ xa = X  + (size_t)(r0 + m) * 128 + 2 * half;
    const float* __restrict__ wb = W  + (size_t)(2 * half) * C + c0 + m;
    const float* __restrict__ rb = Wr + (size_t)(2 * half) * C + c0 + m;

    v8f acc_h[4] = {};
    v8f acc_r[4] = {};

    for (int kb = 0; kb < 128; kb += 4) {
        v2f a;
        a.x = xa[kb];
        a.y = xa[kb + 1];

#pragma unroll
        for (int t = 0; t < 4; ++t) {
            v2f bw, br;
            bw.x = wb[(size_t)kb * C + t * 16];
            bw.y = wb[(size_t)(kb + 1) * C + t * 16];
            br.x = rb[(size_t)kb * C + t * 16];
            br.y = rb[(size_t)(kb + 1) * C + t * 16];

            acc_h[t] = __builtin_amdgcn_wmma_f32_16x16x4_f32(
                false, a, false, bw, (short)0, acc_h[t], false, false);
            acc_r[t] = __builtin_amdgcn_wmma_f32_16x16x4_f32(
                false, a, false, br, (short)0, acc_r[t], false, false);
        }
    }

    float*       __restrict__ hb = H + (size_t)(r0 + 8 * half) * C + c0 + m;
    float*       __restrict__ ob = R + (size_t)(r0 + 8 * half) * C + c0 + m;
    const float* __restrict__ bb = bias + c0 + m;

#pragma unroll
    for (int t = 0; t < 4; ++t) {
        const float b = bb[t * 16];
#pragma unroll
        for (int v = 0; v < 8; ++v) {
            hb[(size_t)v * C + t * 16] = acc_h[t][v];
            ob[(size_t)v * C + t * 16] = acc_r[t][v] + b;
        }
    }
}

// ---------------------------------------------------------------------------
// Per-node attention scalars: s_src[n] = h[n,:]·a_src ; s_dst[n] = h[n,:]·a_dst
// One wave per node; also initializes per-node segment-max/denominator state.
// ---------------------------------------------------------------------------
template <int C>
__global__ __launch_bounds__(256) void gat_node_scalars(
    const float* __restrict__ H,
    const float* __restrict__ a_src,
    const float* __restrict__ a_dst,
    float* __restrict__ s_src,
    float* __restrict__ s_dst,
    unsigned int* __restrict__ umax,
    float* __restrict__ denom,
    int N)
{
    const int node = (blockIdx.x * blockDim.x + threadIdx.x) >> 5;
    const int lane = threadIdx.x & 31;
    if (node >= N) return;

    const float* __restrict__ hrow = H + (size_t)node * C;
    float ss = 0.f, sd = 0.f;
#pragma unroll
    for (int c = 0; c < C / 32; ++c) {
        const float hv = hrow[lane + c * 32];
        ss += hv * a_src[lane + c * 32];
        sd += hv * a_dst[lane + c * 32];
    }
#pragma unroll
    for (int off = 16; off > 0; off >>= 1) {
        ss += __shfl_down(ss, off, 32);
        sd += __shfl_down(sd, off, 32);
    }
    if (lane == 0) {
        s_src[node] = ss;
        s_dst[node] = sd;
        umax[node]  = 0u;     // minimum of the order-preserving unsigned encoding
        denom[node] = 0.f;
    }
}

// Implicit self-loop edge list: e < E0 -> (ei[e], ei[E0+e]); else (e-E0, e-E0)
__device__ __forceinline__ void edge_nodes(const int* __restrict__ ei, int E0,
                                           int e, int& s, int& d)
{
    if (e < E0) { s = ei[e]; d = ei[E0 + e]; }
    else        { s = e - E0; d = e - E0; }
}

__device__ __forceinline__ unsigned int f32_order_encode(float f)
{
    const unsigned int b = __float_as_uint(f);
    return (b & 0x80000000u) ? ~b : (b | 0x80000000u);
}

__device__ __forceinline__ float f32_order_decode(unsigned int u)
{
    const unsigned int b = (u & 0x80000000u) ? (u & 0x7FFFFFFFu) : ~u;
    return __uint_as_float(b);
}

// Pass 1: per-edge LeakyReLU logit + segment max (ordered-uint atomicMax)
__global__ __launch_bounds__(256) void gat_edge_logit(
    const int* __restrict__ ei, int E0, int ET,
    const float* __restrict__ s_src,
    const float* __restrict__ s_dst,
    float* __restrict__ logit,
    unsigned int* __restrict__ umax)
{
    const int e = blockIdx.x * blockDim.x + threadIdx.x;
    if (e >= ET) return;
    int s, d;
    edge_nodes(ei, E0, e, s, d);
    float v = s_src[s] + s_dst[d];
    v = (v > 0.f) ? v : NEG_SLOPE * v;
    logit[e] = v;
    atomicMax(&umax[d], f32_order_encode(v));
}

// Pass 2: ex = exp(logit - max[dst]); denom[dst] += ex   (ex overwrites logit)
__global__ __launch_bounds__(256) void gat_edge_exp(
    const int* __restrict__ ei, int E0, int ET,
    float* __restrict__ logit_ex,
    const unsigned int* __restrict__ umax,
    float* __restrict__ denom)
{
    const int e = blockIdx.x * blockDim.x + threadIdx.x;
    if (e >= ET) return;
    int s, d;
    edge_nodes(ei, E0, e, s, d);
    const float mx = f32_order_decode(umax[d]);
    const float ex = expf(logit_ex[e] - mx);
    logit_ex[e] = ex;
    atomicAdd(&denom[d], ex);
}

// Pass 3: out[dst,:] += (ex/denom[dst]) * h[src,:]. One block per edge,
// one lane per channel (blockDim.x == C) -> coalesced gather + atomics.
template <int C>
__global__ __launch_bounds__(C) void gat_edge_scatter(
    const int* __restrict__ ei, int E0, int ET,
    const float* __restrict__ ex,
    const float* __restrict__ denom,
    const float* __restrict__ H,
    float* __restrict__ out)
{
    const int e = blockIdx.x;
    const int c = threadIdx.x;
    if (e >= ET) return;
    int s, d;
    edge_nodes(ei, E0, e, s, d);
    const float alpha = ex[e] / denom[d];
    atomicAdd(&out[(size_t)d * C + c], alpha * H[(size_t)s * C + c]);
}

__global__ __launch_bounds__(256) void gat_relu_inplace(float* __restrict__ p, int n)
{
    const int i = blockIdx.x * blockDim.x + threadIdx.x;
    if (i < n) {
        const float v = p[i];
        p[i] = v > 0.f ? v : 0.f;
    }
}

// ---------------------------------------------------------------------------
extern "C" void kernel_launch(void* const* d_in, const int* in_sizes, int n_in,
                              void* d_out, int out_size, void* d_ws, size_t ws_size,
                              hipStream_t stream)
{
    const float* x      = (const float*)d_in[0];
    const int*   ei     = (const int*)  d_in[1];
    const float* W1     = (const float*)d_in[2];
    const float* asrc1  = (const float*)d_in[3];
    const float* adst1  = (const float*)d_in[4];
    const float* b1     = (const float*)d_in[5];
    const float* Wres1  = (const float*)d_in[6];
    const float* W2     = (const float*)d_in[7];
    const float* asrc2  = (const float*)d_in[8];
    const float* adst2  = (const float*)d_in[9];
    const float* b2     = (const float*)d_in[10];
    const float* Wres2  = (const float*)d_in[11];
    float* out = (float*)d_out;

    const int N  = in_sizes[0] / 128;   // 50000 (divisible by 16)
    const int E0 = in_sizes[1] / 2;     // 800000
    const int ET = E0 + N;              // + self-loops

    // Workspace layout (256B aligned slices)
    char* w = (char*)d_ws;
    auto take = [&](size_t bytes) -> void* {
        void* p = (void*)w;
        w += (bytes + 255) & ~(size_t)255;
        return p;
    };
    float*        h     = (float*)take((size_t)N * 256 * sizeof(float)); // h1 (C=128) / h2 (C=256)
    float*        out1  = (float*)take((size_t)N * 128 * sizeof(float)); // layer-1 output / layer-2 input
    float*        ssrc  = (float*)take((size_t)N * sizeof(float));
    float*        sdst  = (float*)take((size_t)N * sizeof(float));
    unsigned int* umax  = (unsigned int*)take((size_t)N * sizeof(unsigned int));
    float*        denom = (float*)take((size_t)N * sizeof(float));
    float*        ew    = (float*)take((size_t)ET * sizeof(float));
    (void)ws_size; (void)n_in; (void)out_size;

    const int egrid = (ET + 255) / 256;
    const int ngrid = (N * 32 + 255) / 256;

    // ---------------- Layer 1 (C = 128) ----------------
    gat_gemm_dual_wmma<128><<<dim3(N / 16, 128 / 64), 32, 0, stream>>>(
        x, W1, Wres1, b1, h, out1);
    gat_node_scalars<128><<<ngrid, 256, 0, stream>>>(
        h, asrc1, adst1, ssrc, sdst, umax, denom, N);
    gat_edge_logit<<<egrid, 256, 0, stream>>>(ei, E0, ET, ssrc, sdst, ew, umax);
    gat_edge_exp<<<egrid, 256, 0, stream>>>(ei, E0, ET, ew, umax, denom);
    gat_edge_scatter<128><<<ET, 128, 0, stream>>>(ei, E0, ET, ew, denom, h, out1);
    gat_relu_inplace<<<(N * 128 + 255) / 256, 256, 0, stream>>>(out1, N * 128);

    // ---------------- Layer 2 (C = 256) ----------------
    gat_gemm_dual_wmma<256><<<dim3(N / 16, 256 / 64), 32, 0, stream>>>(
        out1, W2, Wres2, b2, h, out);          // res+bias written straight into d_out
    gat_node_scalars<256><<<ngrid, 256, 0, stream>>>(
        h, asrc2, adst2, ssrc, sdst, umax, denom, N);
    gat_edge_logit<<<egrid, 256, 0, stream>>>(ei, E0, ET, ssrc, sdst, ew, umax);
    gat_edge_exp<<<egrid, 256, 0, stream>>>(ei, E0, ET, ew, umax, denom);
    gat_edge_scatter<256><<<ET, 256, 0, stream>>>(ei, E0, ET, ew, denom, h, out);
}